// Item2CPEncoder_51719996178823
// MI455X (gfx1250) — compile-verified
//
#include <hip/hip_runtime.h>
#include <math.h>

// MI455X / gfx1250, wave32. fp32 basket-attention encoder:
//   H = sigmoid(X @ W1 + b1); logit = H . wq; softmax over items; out = alpha @ X
// Dominant cost: 107 GFLOP fp32 GEMM -> V_WMMA_F32_16X16X4_F32 (keeps fp32 numerics).
// Emb table (102MB) and W1 (1MB) are L2-resident (192MB L2) -> compute-bound.

typedef __attribute__((ext_vector_type(2))) float v2f;
typedef __attribute__((ext_vector_type(4))) float v4f;
typedef __attribute__((ext_vector_type(8))) float v8f;

#define D_MODEL    512
#define MAX_ITEMS  64
#define N_BASKETS  50
#define N_USERS    64
#define LDS_STRIDE 514   // 512 + 2-float pad: A-frag row stride -> +2 banks/row, conflict-free

__global__ __launch_bounds__(256)
void basket_attn_kernel(const int*   __restrict__ inputs,   // [3200, 64]
                        const int*   __restrict__ lengths,  // [3200]
                        const float* __restrict__ emb,      // [50000, 512]
                        const float* __restrict__ W1,       // [512(K), 512(N)] row-major
                        const float* __restrict__ b1,       // [512]
                        const float* __restrict__ wq,       // [512]
                        float*       __restrict__ out)      // [3200, 512]
{
    extern __shared__ float smem[];
    float* Xs      = smem;                                   // 64 x 514
    float* logitsS = smem + MAX_ITEMS * LDS_STRIDE;          // 64
    float* alphaS  = logitsS + MAX_ITEMS;                    // 64

    const int bid  = blockIdx.x;           // basket id in [0, 3200)
    const int tid  = threadIdx.x;          // 0..255 (8 wave32s)
    const int lane = tid & 31;
    const int wave = tid >> 5;
    const int m16  = lane & 15;            // row/col within 16-wide tile
    const int kb   = (lane >> 4) << 1;     // K sub-offset per A/B VGPR layout: 0 or 2

    // ---------- Phase 1: gather 64 embeddings (64x512 f32) into LDS ----------
    const int* items = inputs + bid * MAX_ITEMS;
    #pragma unroll 4
    for (int it = 0; it < 32; ++it) {
        int flat4 = it * 256 + tid;            // float4 index in 64x512
        int row   = flat4 >> 7;                // /128 float4 per row
        int col4  = flat4 & 127;
        int idx   = items[row];
        const float* src = emb + (size_t)idx * D_MODEL + col4 * 4;
        v4f v = *(const v4f*)src;              // global_load_b128, coalesced per row
        float* dst = Xs + row * LDS_STRIDE + col4 * 4;   // 8B aligned
        *(v2f*)(dst)     = v.lo;               // ds_store_b64 x2 (16B-align not guaranteed)
        *(v2f*)(dst + 2) = v.hi;
    }
    if (tid < MAX_ITEMS) logitsS[tid] = 0.0f;
    __syncthreads();

    // ---------- Phase 2: H-tile GEMM via V_WMMA_F32_16X16X4_F32 ----------
    // Wave w owns N-tiles {w, w+8, w+16, w+24} x all 4 M-tiles -> 16 16x16 accumulators.
    const v8f zacc = {0.f,0.f,0.f,0.f,0.f,0.f,0.f,0.f};
    v8f acc[4][4];
    #pragma unroll
    for (int rt = 0; rt < 4; ++rt)
        #pragma unroll
        for (int ci = 0; ci < 4; ++ci)
            acc[rt][ci] = zacc;

    #pragma unroll 1
    for (int k0 = 0; k0 < D_MODEL; k0 += 4) {
        // A frags: 16x4 f32, lane m16 holds X[m, k0+kb], X[m, k0+kb+1]  (ds_load_b64)
        v2f afr[4];
        #pragma unroll
        for (int rt = 0; rt < 4; ++rt)
            afr[rt] = *(const v2f*)(Xs + (rt * 16 + m16) * LDS_STRIDE + k0 + kb);
        // B frags: 4x16 f32, lane m16 holds W1[k0+kb, n], W1[k0+kb+1, n]
        v2f bfr[4];
        #pragma unroll
        for (int ci = 0; ci < 4; ++ci) {
            int n = (wave + ci * 8) * 16 + m16;
            const float* p = W1 + (size_t)(k0 + kb) * D_MODEL + n;
            bfr[ci].x = p[0];
            bfr[ci].y = p[D_MODEL];
        }
        #pragma unroll
        for (int rt = 0; rt < 4; ++rt)
            #pragma unroll
            for (int ci = 0; ci < 4; ++ci)
                acc[rt][ci] = __builtin_amdgcn_wmma_f32_16x16x4_f32(
                    false, afr[rt], false, bfr[ci],
                    (short)0, acc[rt][ci], false, false);
    }

    // ---------- Phase 3: fused sigmoid + dot(wq) -> per-item logits ----------
    // C/D layout: VGPR j = row (tile_m + j) for lanes 0-15, row (tile_m + j + 8) for 16-31;
    // column = tile_n + lane%16.
    #pragma unroll
    for (int rt = 0; rt < 4; ++rt) {
        float part[8];
        #pragma unroll
        for (int j = 0; j < 8; ++j) part[j] = 0.0f;
        #pragma unroll
        for (int ci = 0; ci < 4; ++ci) {
            int n = (wave + ci * 8) * 16 + m16;
            float b1n = b1[n];
            float wqn = wq[n];
            #pragma unroll
            for (int j = 0; j < 8; ++j) {
                float h = acc[rt][ci][j] + b1n;
                part[j] += wqn * (1.0f / (1.0f + __expf(-h)));
            }
        }
        #pragma unroll
        for (int j = 0; j < 8; ++j) {
            float v = part[j];
            #pragma unroll
            for (int m = 8; m >= 1; m >>= 1)
                v += __shfl_xor(v, m, 16);        // reduce across the 16 columns
            if (m16 == 0) {
                int row = rt * 16 + j + ((lane >> 4) << 3);  // +8 for upper half-wave
                atomicAdd(&logitsS[row], v);       // ds_add_f32 across the 8 waves
            }
        }
    }
    __syncthreads();

    // ---------- Phase 4: masked softmax over 64 items (wave 0) ----------
    // bq omitted: softmax is shift-invariant.
    if (wave == 0) {
        int len = lengths[bid];
        float l0 = (lane      < len) ? logitsS[lane]      : -__builtin_inff();
        float l1 = (lane + 32 < len) ? logitsS[lane + 32] : -__builtin_inff();
        float mx = fmaxf(l0, l1);
        #pragma unroll
        for (int m = 16; m >= 1; m >>= 1)
            mx = fmaxf(mx, __shfl_xor(mx, m, 32));
        float e0 = (lane      < len) ? __expf(l0 - mx) : 0.0f;
        float e1 = (lane + 32 < len) ? __expf(l1 - mx) : 0.0f;
        float s = e0 + e1;
        #pragma unroll
        for (int m = 16; m >= 1; m >>= 1)
            s += __shfl_xor(s, m, 32);
        float inv = 1.0f / s;                     // len >= 1 -> s > 0
        alphaS[lane]      = e0 * inv;
        alphaS[lane + 32] = e1 * inv;
    }
    __syncthreads();

    // ---------- Phase 5: out[b, :] = alpha @ X (from LDS, conflict-free) ----------
    float a0 = 0.0f, a1 = 0.0f;
    #pragma unroll 8
    for (int l = 0; l < MAX_ITEMS; ++l) {
        float al = alphaS[l];                     // broadcast
        a0 += al * Xs[l * LDS_STRIDE + tid];
        a1 += al * Xs[l * LDS_STRIDE + tid + 256];
    }
    float* o = out + (size_t)bid * D_MODEL;
    o[tid]       = a0;
    o[tid + 256] = a1;
}

extern "C" void kernel_launch(void* const* d_in, const int* in_sizes, int n_in,
                              void* d_out, int out_size, void* d_ws, size_t ws_size,
                              hipStream_t stream) {
    (void)in_sizes; (void)n_in; (void)out_size; (void)d_ws; (void)ws_size;
    const int*   inputs  = (const int*)d_in[0];   // [64,50,64]
    const int*   lengths = (const int*)d_in[1];   // [64,50]
    const float* emb     = (const float*)d_in[2]; // [50000,512]
    const float* W1      = (const float*)d_in[3]; // [512,512]
    const float* b1      = (const float*)d_in[4]; // [512]
    const float* wq      = (const float*)d_in[5]; // [512]
    // d_in[6] = bq (scalar): unused, softmax is shift-invariant.
    float* out = (float*)d_out;                   // [64,50,512]

    size_t smem = (size_t)(MAX_ITEMS * LDS_STRIDE + 2 * MAX_ITEMS) * sizeof(float); // ~132 KB
    basket_attn_kernel<<<dim3(N_USERS * N_BASKETS), dim3(256), smem, stream>>>(
        inputs, lengths, emb, W1, b1, wq, out);
}